// WC_3745211482216
// MI455X (gfx1250) — compile-verified
//
#include <hip/hip_runtime.h>

typedef float v2f __attribute__((ext_vector_type(2)));
typedef float v8f __attribute__((ext_vector_type(8)));

#define HW   16384
#define CCH  256
#define EPSV 0.001f

// workspace layout (in floats)
#define WS_SUM   0                      // 64*32
#define WS_G     2048                   // 64*32*32
#define WS_MEAN  (2048 + 65536)         // 64*32
#define WS_INVL  (WS_MEAN + 2048)       // 64*32*32
#define WS_AP    (WS_INVL + 65536)      // 8*256*256
#define WS_BIAS  (WS_AP + 524288)       // 8*256

// async copy: 16B per lane, global -> LDS, tracked by ASYNCcnt
__device__ __forceinline__ void async_b128(unsigned lds_off, const float* gptr) {
    asm volatile("global_load_async_to_lds_b128 %0, %1, off"
                 :: "v"(lds_off), "v"((unsigned long long)gptr) : "memory");
}
__device__ __forceinline__ void wait_async_le4() {
    asm volatile("s_wait_asynccnt 0x4" ::: "memory");
}
__device__ __forceinline__ void wait_async_0() {
    asm volatile("s_wait_asynccnt 0x0" ::: "memory");
}

// ---------------------------------------------------------------------------
// Kernel 1: per-group channel sums + X*X^T accumulation via f32 WMMA.
// grid = (8 pixel-chunks, 64 group-instances), block = 128 (4 waves).
// Each wave owns one 16x16 quadrant of the 32x32 Gram matrix.
// Double-buffered async-to-LDS staging (4 async ops/wave/stage).
// ---------------------------------------------------------------------------
__global__ __launch_bounds__(128) void wc_cov_stats(const float* __restrict__ x,
                                                    float* __restrict__ ws) {
    const int gi   = blockIdx.y;            // group instance: b*8 + blk
    const int b    = gi >> 3, blk = gi & 7;
    const int t    = threadIdx.x;
    const int wv   = t >> 5, lane = t & 31, half = lane >> 4, r = lane & 15;
    const int mi   = (wv >> 1) * 16;        // quadrant row base
    const int ni   = (wv & 1) * 16;         // quadrant col base
    const float* xg = x + (size_t)(b * CCH + blk * 32) * HW + (size_t)blockIdx.x * 2048;

    __shared__ __align__(16) float lXc[2][32][68]; // pitch 68: column reads conflict-free

    v8f  acc = {0.f,0.f,0.f,0.f,0.f,0.f,0.f,0.f};
    float ssum = 0.f;
    const int ch = t & 31, q = t >> 5;

    const int srow = t >> 4, sc4 = t & 15;  // this thread's 4 staged rows: srow, srow+8, ...
    auto stage = [&](int sc, int bufi) {
        for (int i = 0; i < 4; ++i) {
            int row = srow + i * 8;
            async_b128((unsigned)(size_t)&lXc[bufi][row][sc4 * 4],
                       xg + (size_t)row * HW + sc * 64 + sc4 * 4);
        }
    };

    stage(0, 0);
    for (int sc = 0; sc < 32; ++sc) {       // 32 sub-tiles of 64 pixels
        const int cur = sc & 1;
        if (sc + 1 < 32) { stage(sc + 1, cur ^ 1); wait_async_le4(); }
        else            { wait_async_0(); }
        __syncthreads();
        for (int j = 0; j < 16; ++j) ssum += lXc[cur][ch][q * 16 + j];
        for (int p = 0; p < 64; p += 4) {
            v2f a, bb;
            a.x  = lXc[cur][mi + r][p + 2*half];  a.y  = lXc[cur][mi + r][p + 2*half + 1];
            bb.x = lXc[cur][ni + r][p + 2*half];  bb.y = lXc[cur][ni + r][p + 2*half + 1];
            acc = __builtin_amdgcn_wmma_f32_16x16x4_f32(false, a, false, bb,
                                                        (short)0, acc, false, false);
        }
        __syncthreads();                     // all waves done reading buf before reuse
    }
    atomicAdd(&ws[WS_SUM + gi * 32 + ch], ssum);
    float* G = ws + WS_G + (size_t)gi * 1024;
    for (int i = 0; i < 8; ++i) {
        int row = mi + half * 8 + i;
        atomicAdd(&G[row * 32 + ni + r], acc[i]);
    }
}

// ---------------------------------------------------------------------------
// Kernel 2: finalize cov, Cholesky, invert L. One wave per group instance.
// ---------------------------------------------------------------------------
__global__ __launch_bounds__(32) void wc_chol(float* __restrict__ ws) {
    const int gi = blockIdx.x;
    const int t  = threadIdx.x;
    __shared__ float sm[32];
    __shared__ float lC[32][33];
    __shared__ float lI[32][33];
    const float inv_hw = 1.0f / (float)HW;

    float m = ws[WS_SUM + gi * 32 + t] * inv_hw;
    sm[t] = m;
    ws[WS_MEAN + gi * 32 + t] = m;
    __syncthreads();

    const float* G = ws + WS_G + (size_t)gi * 1024;
    for (int j = 0; j < 32; ++j) {
        float v = (G[t * 32 + j] * inv_hw - m * sm[j]) * (1.0f - EPSV);
        if (t == j) v += EPSV;
        lC[t][j] = v;
    }
    __syncthreads();
    for (int k = 0; k < 32; ++k) {          // right-looking Cholesky, lane = row
        if (t == k) lC[k][k] = sqrtf(lC[k][k]);
        __syncthreads();
        if (t > k) lC[t][k] /= lC[k][k];
        __syncthreads();
        if (t > k) {
            float lik = lC[t][k];
            for (int j = k + 1; j <= t; ++j) lC[t][j] -= lik * lC[j][k];
        }
        __syncthreads();
    }
    // forward substitution: lane t owns column t of L^{-1}
    for (int i = 0; i < 32; ++i) {
        float s = (i == t) ? 1.0f : 0.0f;
        for (int j = t; j < i; ++j) s -= lC[i][j] * lI[j][t];
        lI[i][t] = (i < t) ? 0.0f : s / lC[i][i];
    }
    float* iv = ws + WS_INVL + (size_t)gi * 1024;
    for (int i = 0; i < 32; ++i) iv[i * 32 + t] = lI[i][t];
}

// ---------------------------------------------------------------------------
// Kernel 3: A'_b = (conv_w + ek[cls]^T) * blockdiag(invL).  grid=(256 d, 8 b).
// ---------------------------------------------------------------------------
__global__ __launch_bounds__(256) void wc_build_A(const float* __restrict__ conv_w,
                                                  const float* __restrict__ ek,
                                                  const int*   __restrict__ cls_id,
                                                  float* __restrict__ ws) {
    const int d = blockIdx.x, b = blockIdx.y, t = threadIdx.x;
    const int cls = cls_id[b];
    __shared__ float lM[256];
    lM[t] = conv_w[d * 256 + t] + ek[(size_t)cls * 65536 + (size_t)t * 256 + d];
    __syncthreads();
    const int blk = t >> 5, j = t & 31;
    const float* il = ws + WS_INVL + (size_t)(b * 8 + blk) * 1024;
    const float* mr = &lM[blk * 32];
    float s = 0.f;
    for (int i = 0; i < 32; ++i) s += mr[i] * il[i * 32 + j];
    ws[WS_AP + ((size_t)(b * 256 + d)) * 256 + t] = s;
}

// ---------------------------------------------------------------------------
// Kernel 4: bias'_b[d] = conv_b[d] + eb[cls][d] - sum_c A'[d][c]*mean[c].
// ---------------------------------------------------------------------------
__global__ __launch_bounds__(256) void wc_build_bias(const float* __restrict__ conv_b,
                                                     const float* __restrict__ eb,
                                                     const int*   __restrict__ cls_id,
                                                     float* __restrict__ ws) {
    const int b = blockIdx.x, d = threadIdx.x;
    const int cls = cls_id[b];
    __shared__ float smn[256];
    smn[d] = ws[WS_MEAN + (b * 8 + (d >> 5)) * 32 + (d & 31)];
    __syncthreads();
    const float* ap = ws + WS_AP + ((size_t)(b * 256 + d)) * 256;
    float s = conv_b[d] + eb[cls * 256 + d];
    for (int c = 0; c < 256; ++c) s -= ap[c] * smn[c];
    ws[WS_BIAS + b * 256 + d] = s;
}

// ---------------------------------------------------------------------------
// Kernel 5: out_b = A'_b (256x256) * x_b (256x16384) + bias'.
// grid = (256 pixel tiles, 4 row tiles, 8 batches), block = 128 (4 waves).
// Wave wv: rows [mBase+16wv, +16), 4 N-subtiles (A-fragment reused x4).
// Double-buffered async-to-LDS staging overlaps loads with WMMA compute.
// ---------------------------------------------------------------------------
__global__ __launch_bounds__(128) void wc_gemm(const float* __restrict__ x,
                                               const float* __restrict__ ws,
                                               float* __restrict__ out) {
    const int pBase = blockIdx.x * 64;
    const int mBase = blockIdx.y * 64;
    const int b     = blockIdx.z;
    const int t     = threadIdx.x;
    const int wv = t >> 5, lane = t & 31, half = lane >> 4, r = lane & 15;

    __shared__ __align__(16) float lA[2][64][20]; // pitch 20: conflict-free column reads
    __shared__ __align__(16) float lX[2][16][72]; // pitch 72: half-wave bank ranges disjoint

    const float* Ap = ws + WS_AP + (size_t)b * 65536;
    const float* xb = x + (size_t)b * CCH * HW;
    const float* bp = ws + WS_BIAS + b * 256;

    v8f acc0, acc1, acc2, acc3;
    for (int i = 0; i < 8; ++i) {
        float bv = bp[mBase + wv * 16 + half * 8 + i];
        acc0[i] = bv; acc1[i] = bv; acc2[i] = bv; acc3[i] = bv;
    }

    const int aRow = t >> 2, aC4 = t & 3;      // A block: 64 x 16 (2 f4 per thread)
    const int xRow = t >> 4, xC4 = t & 15;     // X block: 16 x 64 (2 f4 per thread)
    auto stage = [&](int kB, int bufi) {
        async_b128((unsigned)(size_t)&lA[bufi][aRow][aC4 * 4],
                   Ap + (size_t)(mBase + aRow) * 256 + kB + aC4 * 4);
        async_b128((unsigned)(size_t)&lA[bufi][aRow + 32][aC4 * 4],
                   Ap + (size_t)(mBase + aRow + 32) * 256 + kB + aC4 * 4);
        async_b128((unsigned)(size_t)&lX[bufi][xRow][xC4 * 4],
                   xb + (size_t)(kB + xRow) * HW + pBase + xC4 * 4);
        async_b128((unsigned)(size_t)&lX[bufi][xRow + 8][xC4 * 4],
                   xb + (size_t)(kB + xRow + 8) * HW + pBase + xC4 * 4);
    };

    stage(0, 0);
    for (int kB = 0; kB < 256; kB += 16) {
        const int cur = (kB >> 4) & 1;
        if (kB + 16 < 256) { stage(kB + 16, cur ^ 1); wait_async_le4(); }
        else              { wait_async_0(); }
        __syncthreads();
        for (int ks = 0; ks < 16; ks += 4) {
            v2f a;
            a.x = lA[cur][wv * 16 + r][ks + 2*half];
            a.y = lA[cur][wv * 16 + r][ks + 2*half + 1];
            v2f b0, b1, b2, b3;
            b0.x = lX[cur][ks + 2*half][ 0 + r];  b0.y = lX[cur][ks + 2*half + 1][ 0 + r];
            b1.x = lX[cur][ks + 2*half][16 + r];  b1.y = lX[cur][ks + 2*half + 1][16 + r];
            b2.x = lX[cur][ks + 2*half][32 + r];  b2.y = lX[cur][ks + 2*half + 1][32 + r];
            b3.x = lX[cur][ks + 2*half][48 + r];  b3.y = lX[cur][ks + 2*half + 1][48 + r];
            acc0 = __builtin_amdgcn_wmma_f32_16x16x4_f32(false, a, false, b0, (short)0, acc0, false, false);
            acc1 = __builtin_amdgcn_wmma_f32_16x16x4_f32(false, a, false, b1, (short)0, acc1, false, false);
            acc2 = __builtin_amdgcn_wmma_f32_16x16x4_f32(false, a, false, b2, (short)0, acc2, false, false);
            acc3 = __builtin_amdgcn_wmma_f32_16x16x4_f32(false, a, false, b3, (short)0, acc3, false, false);
        }
        __syncthreads();                       // all waves done reading buf before reuse
    }

    float* ob = out + (size_t)b * CCH * HW;
    for (int i = 0; i < 8; ++i) {
        int d = mBase + wv * 16 + half * 8 + i;
        float* orow = ob + (size_t)d * HW + pBase + r;
        orow[ 0] = acc0[i];
        orow[16] = acc1[i];
        orow[32] = acc2[i];
        orow[48] = acc3[i];
    }
}

// ---------------------------------------------------------------------------
extern "C" void kernel_launch(void* const* d_in, const int* in_sizes, int n_in,
                              void* d_out, int out_size, void* d_ws, size_t ws_size,
                              hipStream_t stream) {
    const float* x   = (const float*)d_in[0];
    const int*   cls = (const int*)  d_in[1];   // class_id (int32 per harness convention)
    const float* ek  = (const float*)d_in[2];
    const float* eb  = (const float*)d_in[3];
    const float* cw  = (const float*)d_in[4];
    const float* cb  = (const float*)d_in[5];
    float* ws  = (float*)d_ws;
    float* out = (float*)d_out;

    // zero the atomic accumulators (sum + Gram); ws is poisoned 0xAA.
    hipMemsetAsync(ws, 0, (size_t)(2048 + 65536) * sizeof(float), stream);

    wc_cov_stats <<<dim3(8, 64),   128, 0, stream>>>(x, ws);
    wc_chol      <<<64,            32,  0, stream>>>(ws);
    wc_build_A   <<<dim3(256, 8),  256, 0, stream>>>(cw, ek, cls, ws);
    wc_build_bias<<<8,             256, 0, stream>>>(cb, eb, cls, ws);
    wc_gemm      <<<dim3(256, 4, 8), 128, 0, stream>>>(x, ws, out);
}